// GaussianMixtureModel_13889924235707
// MI455X (gfx1250) — compile-verified
//
#include <hip/hip_runtime.h>
#include <hip/hip_bf16.h>

// GMM log-prob, N=16384, K=200, D=64, fp32.
// ssq[n,k] = sum_e (x_n . L_k[:,e] - muP[k,e])^2
// WMMA tile: D[e,n] = sum_d L^T[e,d] * x^T[d,n]  (M=e, N=n) so the e-reduction
// is over M (8 acc VGPRs per lane + one cross-half shfl).
// Block = 8 waves: 32 samples x 8 components; each wave: 1 component, 2 n-tiles
// held in registers (8 f32 acc tiles) so each A load feeds 2 WMMAs.

typedef float v2f __attribute__((ext_vector_type(2)));
typedef float v4f __attribute__((ext_vector_type(4)));
typedef float v8f __attribute__((ext_vector_type(8)));

#define GMM_D 64
#define LOG_2PI 1.8378770664093453f
#define XS_STRIDE 68  // 64 + 4 pad -> conflict-free ds_load_b64 across 16 lanes

#if __has_builtin(__builtin_amdgcn_global_load_async_to_lds_b128) && \
    __has_builtin(__builtin_amdgcn_s_wait_asynccnt)
#define USE_ASYNC_LDS 1
typedef int v4i __attribute__((vector_size(16)));
typedef __attribute__((address_space(1))) v4i g_v4i;     // global int4
typedef __attribute__((address_space(3))) v4i lds_v4i;   // LDS int4
#else
#define USE_ASYNC_LDS 0
#endif

// ---------------------------------------------------------------------------
// Prep: muP[k,e] = sum_d means[k,d] * L[k,d,e];  logdet[k] = sum_e log L[k,e,e]
// ---------------------------------------------------------------------------
__global__ __launch_bounds__(64) void gmm_prep_kernel(
    const float* __restrict__ means, const float* __restrict__ prec,
    float* __restrict__ muP, float* __restrict__ logdet)
{
    const int k = blockIdx.x;
    const int e = threadIdx.x;              // 0..63
    const float* Lk = prec + k * (GMM_D * GMM_D);
    const float* mk = means + k * GMM_D;

    float s = 0.0f;
    #pragma unroll 8
    for (int d = 0; d < GMM_D; ++d)
        s = fmaf(mk[d], Lk[d * GMM_D + e], s);
    muP[k * GMM_D + e] = s;

    __shared__ float red[GMM_D];
    red[e] = __logf(Lk[e * GMM_D + e]);
    __syncthreads();
    if (e == 0) {
        float t = 0.0f;
        #pragma unroll
        for (int i = 0; i < GMM_D; ++i) t += red[i];
        logdet[k] = t;
    }
}

// ---------------------------------------------------------------------------
// Main kernel: grid (K/8, N/32), block 256.
// ---------------------------------------------------------------------------
__global__ __launch_bounds__(256) void gmm_wmma_kernel(
    const float* __restrict__ x, const float* __restrict__ prec,
    const float* __restrict__ muP, const float* __restrict__ logdet,
    float* __restrict__ out, int Kcomp)
{
    __shared__ float xs[32 * XS_STRIDE];

    const int n0    = blockIdx.y * 32;
    const int kbase = blockIdx.x * 8;
    const int tid   = threadIdx.x;

    // ---- Stage x tile [32 x 64] -> LDS (row stride 68 floats) ----
    // 32 rows x 16 x b128 chunks = 512 chunks, 2 per thread.
#if USE_ASYNC_LDS
    {
        #pragma unroll
        for (int j = 0; j < 2; ++j) {
            const int idx = tid + j * 256;
            const int row = idx >> 4;       // 0..31
            const int seg = idx & 15;       // 0..15
            g_v4i*   src = (g_v4i*)(x + (n0 + row) * GMM_D + seg * 4);
            lds_v4i* dst = (lds_v4i*)(&xs[row * XS_STRIDE + seg * 4]);
            __builtin_amdgcn_global_load_async_to_lds_b128(src, dst, 0, 0);
        }
        __builtin_amdgcn_s_wait_asynccnt(0);
    }
#else
    {
        #pragma unroll
        for (int j = 0; j < 2; ++j) {
            const int idx = tid + j * 256;
            const int row = idx >> 4;
            const int seg = idx & 15;
            v4f v = *(const v4f*)(x + (n0 + row) * GMM_D + seg * 4);
            *(v4f*)(&xs[row * XS_STRIDE + seg * 4]) = v;
        }
    }
#endif
    __syncthreads();

    const int wave = tid >> 5;              // 0..7
    const int lane = tid & 31;
    const int l15  = lane & 15;
    const int half = lane >> 4;             // 0 or 1
    const int k    = kbase + wave;

    const float* Lk    = prec + k * (GMM_D * GMM_D);  // [d][e]
    const float* xrow0 = &xs[l15 * XS_STRIDE];                 // n-tile 0
    const float* xrow1 = &xs[(16 + l15) * XS_STRIDE];          // n-tile 1
    const int dadd = 2 * half;              // K-slots {0,1} low half, {2,3} high

    v8f a00 = {}, a01 = {}, a02 = {}, a03 = {};   // n-tile 0, e-tiles 0..3
    v8f a10 = {}, a11 = {}, a12 = {}, a13 = {};   // n-tile 1, e-tiles 0..3

    #pragma unroll
    for (int d0 = 0; d0 < GMM_D; d0 += 4) {
        // B operands: x[n, d0+dadd .. +1] for both n-tiles
        v2f b0 = *(const v2f*)(xrow0 + d0 + dadd);
        v2f b1 = *(const v2f*)(xrow1 + d0 + dadd);

        // A operands: av[j] = L[d0+dadd+j][etile*16 + l15] (shared by 2 WMMAs each)
        const float* La = Lk + (d0 + dadd) * GMM_D + l15;
        v2f av0, av1, av2, av3;
        av0.x = La[0];   av0.y = La[64];
        av1.x = La[16];  av1.y = La[80];
        av2.x = La[32];  av2.y = La[96];
        av3.x = La[48];  av3.y = La[112];

        a00 = __builtin_amdgcn_wmma_f32_16x16x4_f32(false, av0, false, b0, (short)0, a00, false, false);
        a10 = __builtin_amdgcn_wmma_f32_16x16x4_f32(false, av0, false, b1, (short)0, a10, false, false);
        a01 = __builtin_amdgcn_wmma_f32_16x16x4_f32(false, av1, false, b0, (short)0, a01, false, false);
        a11 = __builtin_amdgcn_wmma_f32_16x16x4_f32(false, av1, false, b1, (short)0, a11, false, false);
        a02 = __builtin_amdgcn_wmma_f32_16x16x4_f32(false, av2, false, b0, (short)0, a02, false, false);
        a12 = __builtin_amdgcn_wmma_f32_16x16x4_f32(false, av2, false, b1, (short)0, a12, false, false);
        a03 = __builtin_amdgcn_wmma_f32_16x16x4_f32(false, av3, false, b0, (short)0, a03, false, false);
        a13 = __builtin_amdgcn_wmma_f32_16x16x4_f32(false, av3, false, b1, (short)0, a13, false, false);
    }

    // Epilogue: acc VGPR r holds e = etile*16 + r + 8*half, n = l15 (per n-tile).
    const float* mPk = muP + k * GMM_D;
    float p0 = 0.0f, p1 = 0.0f;
    #pragma unroll
    for (int r = 0; r < 8; ++r) {
        const float m0 = mPk[ 0 + 8 * half + r];
        const float m1 = mPk[16 + 8 * half + r];
        const float m2 = mPk[32 + 8 * half + r];
        const float m3 = mPk[48 + 8 * half + r];
        float d0v = a00[r] - m0, d1v = a01[r] - m1, d2v = a02[r] - m2, d3v = a03[r] - m3;
        p0 = fmaf(d0v, d0v, p0); p0 = fmaf(d1v, d1v, p0);
        p0 = fmaf(d2v, d2v, p0); p0 = fmaf(d3v, d3v, p0);
        float e0v = a10[r] - m0, e1v = a11[r] - m1, e2v = a12[r] - m2, e3v = a13[r] - m3;
        p1 = fmaf(e0v, e0v, p1); p1 = fmaf(e1v, e1v, p1);
        p1 = fmaf(e2v, e2v, p1); p1 = fmaf(e3v, e3v, p1);
    }

    // Cross-half reduction (wave32): lane n adds the M=8..15 half.
    float s0 = p0 + __shfl_xor(p0, 16, 32);
    float s1 = p1 + __shfl_xor(p1, 16, 32);

    if (lane < 16) {
        const float cterm = logdet[k] - 0.5f * (float)GMM_D * LOG_2PI;
        out[(n0 + lane) * Kcomp + k]      = fmaf(-0.5f, s0, cterm);
        out[(n0 + 16 + lane) * Kcomp + k] = fmaf(-0.5f, s1, cterm);
    }
}

// ---------------------------------------------------------------------------
extern "C" void kernel_launch(void* const* d_in, const int* in_sizes, int n_in,
                              void* d_out, int out_size, void* d_ws, size_t ws_size,
                              hipStream_t stream) {
    const float* x     = (const float*)d_in[0];   // [N, 64]
    const float* means = (const float*)d_in[1];   // [K, 64]
    const float* prec  = (const float*)d_in[2];   // [K, 64, 64]
    float* out = (float*)d_out;                   // [N, K]

    const int N = in_sizes[0] / GMM_D;            // 16384
    const int K = in_sizes[1] / GMM_D;            // 200

    float* muP    = (float*)d_ws;                 // K*64 floats
    float* logdet = muP + K * GMM_D;              // K floats

    gmm_prep_kernel<<<K, 64, 0, stream>>>(means, prec, muP, logdet);

    dim3 grid(K / 8, N / 32);
    gmm_wmma_kernel<<<grid, 256, 0, stream>>>(x, prec, muP, logdet, out, K);
}